// LocalPearsonDepthLoss_33517924778561
// MI455X (gfx1250) — compile-verified
//
#include <hip/hip_runtime.h>
#include <hip/hip_bf16.h>

// CDNA5 / gfx1250, wave32. WMMA f32 16x16x4 used as an exact f32 reduction tree.

typedef __attribute__((ext_vector_type(2))) float v2f;
typedef __attribute__((ext_vector_type(8))) float v8f;

#define BOX_P     128
#define IMG_W     4096
#define BOX_ELEMS (BOX_P * BOX_P)   // 16384
#define EPS_F     1e-6f

// One workgroup (8 wave32s) per box. Five WMMA accumulator chains compute
// Sum(p), Sum(g), Sum(p^2), Sum(g^2), Sum(p*g) over the 128x128 patch.
__global__ __launch_bounds__(256) void pearson_box_kernel(
    const float* __restrict__ P, const float* __restrict__ G,
    const int* __restrict__ x0, const int* __restrict__ y0,
    float* __restrict__ box_loss)
{
    const int box  = blockIdx.x;
    const int tid  = threadIdx.x;
    const int wave = tid >> 5;
    const int lane = tid & 31;

    const long base = (long)x0[box] * IMG_W + (long)y0[box];

    v8f aP  = {};
    v8f aG  = {};
    v8f aPP = {};
    v8f aGG = {};
    v8f aPG = {};
    v2f ones; ones.x = 1.0f; ones.y = 1.0f;   // B matrix: all ones -> row sums

    // 16384 elements / (8 waves * 64 per WMMA step) = 32 iterations.
    // Each 64-element chunk is half of one box row -> coalesced 128B loads.
#pragma unroll 4
    for (int i = 0; i < 32; ++i) {
        const int  chunk = (i * 8 + wave) * 64;      // multiple of 64, within one row
        const int  row   = chunk >> 7;               // chunk / 128
        const int  col   = chunk & 127;
        const long rb    = base + (long)row * IMG_W + col;

        const float p0 = P[rb + lane];
        const float p1 = P[rb + lane + 32];
        const float g0 = G[rb + lane];
        const float g1 = G[rb + lane + 32];

        v2f ap;  ap.x  = p0;      ap.y  = p1;
        v2f ag;  ag.x  = g0;      ag.y  = g1;
        v2f app; app.x = p0 * p0; app.y = p1 * p1;
        v2f agg; agg.x = g0 * g0; agg.y = g1 * g1;
        v2f apg; apg.x = p0 * g0; apg.y = p1 * g1;

        // D = A * ones(4x16) + C : every column of D holds the 16 row-sums of A.
        aP  = __builtin_amdgcn_wmma_f32_16x16x4_f32(false, ap,  false, ones, (short)0, aP,  false, false);
        aG  = __builtin_amdgcn_wmma_f32_16x16x4_f32(false, ag,  false, ones, (short)0, aG,  false, false);
        aPP = __builtin_amdgcn_wmma_f32_16x16x4_f32(false, app, false, ones, (short)0, aPP, false, false);
        aGG = __builtin_amdgcn_wmma_f32_16x16x4_f32(false, agg, false, ones, (short)0, aGG, false, false);
        aPG = __builtin_amdgcn_wmma_f32_16x16x4_f32(false, apg, false, ones, (short)0, aPG, false, false);
    }

    // C/D layout: VGPR r at lane n(0..15) = D[r][n]; at lane n+16 = D[r+8][n].
    // Sum of the 8 components per lane = half the rows for one column; adding
    // the paired lane (xor 16) gives the full matrix-total (columns identical).
    float sP = 0.f, sG = 0.f, sPP = 0.f, sGG = 0.f, sPG = 0.f;
#pragma unroll
    for (int r = 0; r < 8; ++r) {
        sP  += aP[r];
        sG  += aG[r];
        sPP += aPP[r];
        sGG += aGG[r];
        sPG += aPG[r];
    }
    sP  += __shfl_xor(sP,  16, 32);
    sG  += __shfl_xor(sG,  16, 32);
    sPP += __shfl_xor(sPP, 16, 32);
    sGG += __shfl_xor(sGG, 16, 32);
    sPG += __shfl_xor(sPG, 16, 32);

    __shared__ float red[8][5];
    if (lane == 0) {
        red[wave][0] = sP;
        red[wave][1] = sG;
        red[wave][2] = sPP;
        red[wave][3] = sGG;
        red[wave][4] = sPG;
    }
    __syncthreads();

    if (tid == 0) {
        float Sp = 0.f, Sg = 0.f, Spp = 0.f, Sgg = 0.f, Spg = 0.f;
#pragma unroll
        for (int w = 0; w < 8; ++w) {
            Sp  += red[w][0];
            Sg  += red[w][1];
            Spp += red[w][2];
            Sgg += red[w][3];
            Spg += red[w][4];
        }
        const float n   = (float)BOX_ELEMS;
        const float cpp = Spp - Sp * Sp / n;     // Sum((p-mean_p)^2)
        const float cgg = Sgg - Sg * Sg / n;     // Sum((g-mean_g)^2)
        const float cpg = Spg - Sp * Sg / n;     // Sum((p-mean_p)(g-mean_g))
        const float stdp = sqrtf(cpp / (n - 1.0f));   // ddof=1, matches torch.std
        const float stdg = sqrtf(cgg / (n - 1.0f));
        const float co   = (cpg / n) / ((stdp + EPS_F) * (stdg + EPS_F));
        box_loss[box] = 1.0f - co;
    }
}

// Deterministic fixed-order final reduction: out = mean(box_loss[0..n))
__global__ __launch_bounds__(512) void pearson_final_reduce(
    const float* __restrict__ box_loss, float* __restrict__ out, int n)
{
    __shared__ float s[512];
    const int t = threadIdx.x;
    s[t] = (t < n) ? box_loss[t] : 0.0f;
    __syncthreads();
#pragma unroll
    for (int k = 256; k > 0; k >>= 1) {
        if (t < k) s[t] += s[t + k];
        __syncthreads();
    }
    if (t == 0) out[0] = s[0] / (float)n;
}

extern "C" void kernel_launch(void* const* d_in, const int* in_sizes, int n_in,
                              void* d_out, int out_size, void* d_ws, size_t ws_size,
                              hipStream_t stream) {
    const float* depth_pred = (const float*)d_in[0];
    const float* depth_gt   = (const float*)d_in[1];
    const int*   x0         = (const int*)d_in[2];
    const int*   y0         = (const int*)d_in[3];
    // d_in[4] = box_p (==128, baked into BOX_P)

    const int n_corr = in_sizes[2];          // 512 boxes
    float* box_loss = (float*)d_ws;          // n_corr floats of scratch

    pearson_box_kernel<<<n_corr, 256, 0, stream>>>(depth_pred, depth_gt, x0, y0, box_loss);
    pearson_final_reduce<<<1, 512, 0, stream>>>(box_loss, (float*)d_out, n_corr);
}